// CrystalGCNWithExtraFeatures_15865609191627
// MI455X (gfx1250) — compile-verified
//
#include <hip/hip_runtime.h>
#include <cstdint>
#include <cstddef>

// ---------------------------------------------------------------------------
// 3-layer GCN (GCNConv x3 + ReLU + log_softmax) for MI455X (gfx1250, wave32)
// GEMMs: V_WMMA_F32_16X16X4_F32, templated (K, FOUT, RELU) -> immediate
// strides, no bounds guards (N = 50000 = 3125*16 exactly), zero-padded W3.
// Aggregation: wave/edge, scalarized indices, float4 gather + hw f32 atomics.
// ---------------------------------------------------------------------------

typedef __attribute__((ext_vector_type(2))) float v2f;
typedef __attribute__((ext_vector_type(8))) float v8f;

#define GCN_N_NODES 50000
#define GCN_N_EDGES 800000
#define GCN_F_IN    128
#define GCN_F_HID   256
#define GCN_F_OUT   10
#define GCN_F_OUT_P 16   // layer-3 W/out padded to one 16-col WMMA tile

static_assert(GCN_N_NODES % 16 == 0, "row tiling assumes N % 16 == 0");

// ---------------- degree / normalization ----------------------------------

__global__ void deg_init_kernel(float* __restrict__ deg, int n) {
  int i = blockIdx.x * blockDim.x + threadIdx.x;
  if (i < n) deg[i] = 1.0f;  // self-loop contributes 1 to every node's degree
}

__global__ void deg_acc_kernel(const long long* __restrict__ dst,
                               float* __restrict__ deg, int e) {
  int i = blockIdx.x * blockDim.x + threadIdx.x;
  if (i < e) unsafeAtomicAdd(&deg[dst[i]], 1.0f);
}

__global__ void deg_rsqrt_kernel(float* __restrict__ deg, int n) {
  int i = blockIdx.x * blockDim.x + threadIdx.x;
  if (i < n) deg[i] = rsqrtf(deg[i]);  // deg >= 1 always (self-loops)
}

// ---------------- bias init of aggregation buffer --------------------------

__global__ void bias_init_kernel(float* __restrict__ out,
                                 const float* __restrict__ b,
                                 int n, int F, int Freal) {
  int i = blockIdx.x * blockDim.x + threadIdx.x;
  if (i < n * F) {
    int f = i % F;
    out[i] = (f < Freal) ? b[f] : 0.0f;
  }
}

// ---------------- pack W3 [256x10] into zero-padded [256x16] ----------------

__global__ void pack_w3_kernel(const float* __restrict__ W3,
                               float* __restrict__ Wp) {
  int i = blockIdx.x * blockDim.x + threadIdx.x;
  if (i < GCN_F_HID * GCN_F_OUT_P) {
    int r = i >> 4, c = i & 15;
    Wp[i] = (c < GCN_F_OUT) ? W3[r * GCN_F_OUT + c] : 0.0f;
  }
}

// ---------------- WMMA f32 GEMM: Out[N x FOUT] = A[N x K] @ W --------------
// Block = 256 threads = 8 waves, one 16-row tile per block. A staged in LDS
// via float4 (b128) loads/stores; +4-float row pad keeps rows 16B-aligned
// and makes the ds_load_2addr_b64 A-fragment reads bank-conflict-free.
// A fragment (16x4 f32): lane<16 holds K=k,k+1 ; lane>=16 holds K=k+2,k+3.
// Each wave owns col tiles {waveId, waveId+8}: two independent WMMA chains,
// A fragment reused across both B tiles.
// C/D layout: acc[i] = row (i + (lane<16 ? 0 : 8)), col = lane&15.

template <int K, int FOUT, bool RELU>
__launch_bounds__(256)
__global__ void gemm_wmma_kernel(const float* __restrict__ A,
                                 const float* __restrict__ W,   // [K][FOUT]
                                 float* __restrict__ Out) {     // [N][FOUT]
  __shared__ float As[16][K + 4];
  const int tid  = threadIdx.x;
  const int row0 = blockIdx.x * 16;

  // Cooperative A-tile load (vectorized), ReLU fused for layers 2/3.
  {
    const int r  = tid >> 4;            // 16 rows, 16 threads per row
    const int c0 = tid & 15;            // float4 column index
    const float4* Arow = (const float4*)(A + (size_t)(row0 + r) * K);
#pragma unroll
    for (int c = c0; c < K / 4; c += 16) {
      float4 v = Arow[c];
      if (RELU) {
        v.x = fmaxf(v.x, 0.0f); v.y = fmaxf(v.y, 0.0f);
        v.z = fmaxf(v.z, 0.0f); v.w = fmaxf(v.w, 0.0f);
      }
      *(float4*)&As[r][c * 4] = v;      // rows 16B-aligned (K+4 pad)
    }
  }
  __syncthreads();

  const int lane   = tid & 31;
  const int waveId = tid >> 5;
  constexpr int nTiles = FOUT / 16;
  const int t0 = waveId;
  const int t1 = waveId + 8;
  if (t0 >= nTiles) return;  // wave-uniform

  const int m     = lane & 15;
  const int kb    = (lane < 16) ? 0 : 2;
  const int mbase = (lane < 16) ? 0 : 8;
  const float* As_m = &As[m][kb];
  const float* Wp0  = W + kb * FOUT + t0 * 16 + m;

  v8f acc0 = {};
  if (t1 < nTiles) {
    const float* Wp1 = W + kb * FOUT + t1 * 16 + m;
    v8f acc1 = {};
#pragma unroll 4
    for (int k = 0; k < K; k += 4) {
      v2f a;  a.x  = As_m[k];  a.y  = As_m[k + 1];
      v2f b0; b0.x = Wp0[0];   b0.y = Wp0[FOUT];
      v2f b1; b1.x = Wp1[0];   b1.y = Wp1[FOUT];
      acc0 = __builtin_amdgcn_wmma_f32_16x16x4_f32(
          false, a, false, b0, (short)0, acc0, false, false);
      acc1 = __builtin_amdgcn_wmma_f32_16x16x4_f32(
          false, a, false, b1, (short)0, acc1, false, false);
      Wp0 += 4 * FOUT;
      Wp1 += 4 * FOUT;
    }
    float* o0 = Out + (size_t)(row0 + mbase) * FOUT + t0 * 16 + m;
    float* o1 = Out + (size_t)(row0 + mbase) * FOUT + t1 * 16 + m;
#pragma unroll
    for (int i = 0; i < 8; ++i) {       // rows exact: no guards, clause-able
      o0[i * FOUT] = acc0[i];
      o1[i * FOUT] = acc1[i];
    }
  } else {
#pragma unroll 4
    for (int k = 0; k < K; k += 4) {
      v2f a;  a.x  = As_m[k];  a.y  = As_m[k + 1];
      v2f b0; b0.x = Wp0[0];   b0.y = Wp0[FOUT];
      acc0 = __builtin_amdgcn_wmma_f32_16x16x4_f32(
          false, a, false, b0, (short)0, acc0, false, false);
      Wp0 += 4 * FOUT;
    }
    float* o0 = Out + (size_t)(row0 + mbase) * FOUT + t0 * 16 + m;
#pragma unroll
    for (int i = 0; i < 8; ++i) o0[i * FOUT] = acc0[i];
  }
}

// ---------------- edge aggregation: out[dst] += dinv[src]*dinv[dst]*xw[src] -
// One wave per edge (or self-loop). src/dst are wave-uniform -> readfirstlane
// scalarizes index math and dinv loads. Gather as float4 (b128), scatter via
// hw global_atomic_add_f32 (L2-resident: 51MB feature buffer << 192MB L2).

template <int F>
__launch_bounds__(256)
__global__ void agg_kernel(const float* __restrict__ xw,
                           const long long* __restrict__ src,
                           const long long* __restrict__ dst,
                           const float* __restrict__ dinv,
                           float* __restrict__ out,
                           int nEdges, int nNodes) {
  const int wave = (blockIdx.x * blockDim.x + threadIdx.x) >> 5;
  const int lane = threadIdx.x & 31;
  if (wave >= nEdges + nNodes) return;

  int s, d;
  float nrm;
  if (wave < nEdges) {                 // wave-uniform branch
    s = __builtin_amdgcn_readfirstlane((int)src[wave]);
    d = __builtin_amdgcn_readfirstlane((int)dst[wave]);
    nrm = dinv[s] * dinv[d];
  } else {                             // self-loop i -> i
    s = d = wave - nEdges;
    const float di = dinv[s];
    nrm = di * di;
  }

  const float4* xs = (const float4*)(xw + (size_t)s * F);
  float*        od = out + (size_t)d * F;
#pragma unroll
  for (int f = lane; f < F / 4; f += 32) {
    const float4 v = xs[f];
    float* p = od + f * 4;
    unsafeAtomicAdd(p + 0, nrm * v.x);
    unsafeAtomicAdd(p + 1, nrm * v.y);
    unsafeAtomicAdd(p + 2, nrm * v.z);
    unsafeAtomicAdd(p + 3, nrm * v.w);
  }
}

// ---------------- row log-softmax over 10 classes ---------------------------

__global__ void logsoftmax_kernel(const float* __restrict__ h,
                                  float* __restrict__ out, int n) {
  int i = blockIdx.x * blockDim.x + threadIdx.x;
  if (i >= n) return;
  const float4* r4 = (const float4*)(h + (size_t)i * GCN_F_OUT_P);
  float v[12];
  const float4 a = r4[0], b = r4[1], c = r4[2];  // 12 >= 10 classes
  v[0] = a.x; v[1] = a.y; v[2]  = a.z; v[3]  = a.w;
  v[4] = b.x; v[5] = b.y; v[6]  = b.z; v[7]  = b.w;
  v[8] = c.x; v[9] = c.y;
  float mx = -3.402823466e38f;
#pragma unroll
  for (int f = 0; f < GCN_F_OUT; ++f) mx = fmaxf(mx, v[f]);
  float s = 0.0f;
#pragma unroll
  for (int f = 0; f < GCN_F_OUT; ++f) s += expf(v[f] - mx);
  const float lse = logf(s) + mx;
  float* o = out + (size_t)i * GCN_F_OUT;
#pragma unroll
  for (int f = 0; f < GCN_F_OUT; ++f) o[f] = v[f] - lse;
}

// ---------------------------------------------------------------------------

extern "C" void kernel_launch(void* const* d_in, const int* in_sizes, int n_in,
                              void* d_out, int out_size, void* d_ws, size_t ws_size,
                              hipStream_t stream) {
  (void)in_sizes; (void)n_in; (void)out_size; (void)ws_size;

  const float*     x    = (const float*)d_in[0];
  const long long* eidx = (const long long*)d_in[1];
  const float*     W1   = (const float*)d_in[2];
  const float*     b1   = (const float*)d_in[3];
  const float*     W2   = (const float*)d_in[4];
  const float*     b2   = (const float*)d_in[5];
  const float*     W3   = (const float*)d_in[6];
  const float*     b3   = (const float*)d_in[7];
  float*           out  = (float*)d_out;

  const long long* srcI = eidx;                 // edge_index[0]
  const long long* dstI = eidx + GCN_N_EDGES;   // edge_index[1]

  // workspace carve-out
  char* ws = (char*)d_ws;
  size_t off = 0;
  auto carve = [&](size_t bytes) -> void* {
    void* p = ws + off;
    off = (off + bytes + 255) & ~(size_t)255;
    return p;
  };
  float* dinv  = (float*)carve((size_t)GCN_N_NODES * sizeof(float));
  float* bufA  = (float*)carve((size_t)GCN_N_NODES * GCN_F_HID * sizeof(float));
  float* bufB  = (float*)carve((size_t)GCN_N_NODES * GCN_F_HID * sizeof(float));
  float* bufC  = (float*)carve((size_t)GCN_N_NODES * GCN_F_OUT_P * sizeof(float));
  float* W3pad = (float*)carve((size_t)GCN_F_HID * GCN_F_OUT_P * sizeof(float));

  const dim3 blk(256);
  const int  mTiles = GCN_N_NODES / 16;         // 3125 (exact)
  const long long waves = (long long)GCN_N_EDGES + GCN_N_NODES;
  const int  aggBlocks = (int)((waves * 32 + 255) / 256);
  const int  gN   = (GCN_N_NODES + 255) / 256;
  const int  gE   = (GCN_N_EDGES + 255) / 256;
  const int  gHid = ((GCN_N_NODES * GCN_F_HID) + 255) / 256;
  const int  gOut = ((GCN_N_NODES * GCN_F_OUT_P) + 255) / 256;
  const int  gW3  = ((GCN_F_HID * GCN_F_OUT_P) + 255) / 256;

  // ---- degrees -> dinv = rsqrt(deg) ; pack padded W3 ----
  deg_init_kernel <<<gN, blk, 0, stream>>>(dinv, GCN_N_NODES);
  deg_acc_kernel  <<<gE, blk, 0, stream>>>(dstI, dinv, GCN_N_EDGES);
  deg_rsqrt_kernel<<<gN, blk, 0, stream>>>(dinv, GCN_N_NODES);
  pack_w3_kernel  <<<gW3, blk, 0, stream>>>(W3, W3pad);

  // ---- layer 1: xw = x @ W1 ; h1 = scatter(norm * xw) + b1 ----
  gemm_wmma_kernel<GCN_F_IN, GCN_F_HID, false><<<mTiles, blk, 0, stream>>>(
      x, W1, bufA);
  bias_init_kernel<<<gHid, blk, 0, stream>>>(bufB, b1, GCN_N_NODES, GCN_F_HID, GCN_F_HID);
  agg_kernel<GCN_F_HID><<<aggBlocks, blk, 0, stream>>>(
      bufA, srcI, dstI, dinv, bufB, GCN_N_EDGES, GCN_N_NODES);

  // ---- layer 2: xw = relu(h1) @ W2 ; h2 = scatter + b2 ----
  gemm_wmma_kernel<GCN_F_HID, GCN_F_HID, true><<<mTiles, blk, 0, stream>>>(
      bufB, W2, bufA);
  bias_init_kernel<<<gHid, blk, 0, stream>>>(bufB, b2, GCN_N_NODES, GCN_F_HID, GCN_F_HID);
  agg_kernel<GCN_F_HID><<<aggBlocks, blk, 0, stream>>>(
      bufA, srcI, dstI, dinv, bufB, GCN_N_EDGES, GCN_N_NODES);

  // ---- layer 3: xw = relu(h2) @ W3pad (cols 10..15 zero) ----
  gemm_wmma_kernel<GCN_F_HID, GCN_F_OUT_P, true><<<mTiles, blk, 0, stream>>>(
      bufB, W3pad, bufA);
  bias_init_kernel<<<gOut, blk, 0, stream>>>(bufC, b3, GCN_N_NODES, GCN_F_OUT_P, GCN_F_OUT);
  agg_kernel<GCN_F_OUT_P><<<aggBlocks, blk, 0, stream>>>(
      bufA, srcI, dstI, dinv, bufC, GCN_N_EDGES, GCN_N_NODES);

  // ---- log_softmax ----
  logsoftmax_kernel<<<gN, blk, 0, stream>>>(bufC, out, GCN_N_NODES);
}